// HeteroGraphSAGE_9328668967670
// MI455X (gfx1250) — compile-verified
//
#include <hip/hip_runtime.h>

#define H 128

typedef __attribute__((ext_vector_type(2))) float v2f;
typedef __attribute__((ext_vector_type(8))) float v8f;

// V_WMMA_F32_16X16X4_F32: D = A(16x4) * B(4x16) + C, full fp32 (exact vs reference).
__device__ __forceinline__ v8f wmma4(v2f a, v2f b, v8f c) {
    return __builtin_amdgcn_wmma_f32_16x16x4_f32(false, a, false, b, (short)0, c,
                                                 false, false);
}

// ---------------------------------------------------------------------------
// Input projection: out[M,128] = A[M,K] @ W[K,128] + bias   (K arbitrary)
// One wave per 16-row panel; 8 column tiles (full N=128) per wave.
// Row guard is store-side only: invalid A rows are clamped to row 0 (in-bounds)
// and only pollute C rows that are never stored (A row m -> C row m in WMMA).
// ---------------------------------------------------------------------------
__global__ __launch_bounds__(256) void proj_kernel(
    const float* __restrict__ A, const float* __restrict__ W,
    const float* __restrict__ bias, float* __restrict__ out, int M, int K) {
    const int lane = threadIdx.x & 31;
    const int wave = threadIdx.x >> 5;
    const int half = lane >> 4;     // lanes 16-31 hold K+2,K+3
    const int l16  = lane & 15;
    const int rowA = blockIdx.x * 128 + wave * 16 + l16;
    const int rowC = rowA < M ? rowA : 0;           // clamp: loads always in-bounds
    const float* arow = A + (size_t)rowC * K;

    v8f acc[8];
    #pragma unroll
    for (int n = 0; n < 8; ++n) { v8f z = {}; acc[n] = z; }

    const int Kmain = K & ~3;                       // unguarded main body
    for (int k0 = 0; k0 < Kmain; k0 += 4) {
        const int ka = k0 + 2 * half;
        v2f a;
        a[0] = arow[ka];
        a[1] = arow[ka + 1];
        #pragma unroll
        for (int n = 0; n < 8; ++n) {
            const int cb = n * 16 + l16;
            v2f b;
            b[0] = W[(size_t)ka       * H + cb];
            b[1] = W[(size_t)(ka + 1) * H + cb];
            acc[n] = wmma4(a, b, acc[n]);
        }
    }
    if (Kmain < K) {                                // single guarded tail step
        const int ka = Kmain + 2 * half;
        v2f a;
        a[0] = (ka     < K) ? arow[ka]     : 0.f;
        a[1] = (ka + 1 < K) ? arow[ka + 1] : 0.f;
        #pragma unroll
        for (int n = 0; n < 8; ++n) {
            const int cb = n * 16 + l16;
            v2f b;
            b[0] = (ka     < K) ? W[(size_t)ka       * H + cb] : 0.f;
            b[1] = (ka + 1 < K) ? W[(size_t)(ka + 1) * H + cb] : 0.f;
            acc[n] = wmma4(a, b, acc[n]);
        }
    }
    const int rbase = blockIdx.x * 128 + wave * 16 + 8 * half;
    #pragma unroll
    for (int n = 0; n < 8; ++n) {
        const int cb = n * 16 + l16;
        const float bv = bias[cb];
        #pragma unroll
        for (int v = 0; v < 8; ++v) {
            const int r = rbase + v;
            if (r < M) out[(size_t)r * H + cb] = acc[n][v] + bv;
        }
    }
}

// ---------------------------------------------------------------------------
// Edge in-degree counts (float, matches reference segment_sum of ones).
// ---------------------------------------------------------------------------
__global__ __launch_bounds__(256) void count_kernel(
    const int* __restrict__ dst, int E, float* __restrict__ cnt) {
    const int i = blockIdx.x * blockDim.x + threadIdx.x;
    if (i < E) atomicAdd(&cnt[dst[i]], 1.0f);
}

// ---------------------------------------------------------------------------
// Segment scatter-add: seg[dst[e]] += z[src[e]]  (one wave per edge,
// float4 coalesced gather, 4 f32 atomics per lane -> L2 atomic units).
// ---------------------------------------------------------------------------
__global__ __launch_bounds__(256) void scatter_kernel(
    const float* __restrict__ z, const int* __restrict__ src,
    const int* __restrict__ dst, int E, float* __restrict__ seg) {
    const int e = blockIdx.x * (blockDim.x >> 5) + (threadIdx.x >> 5);
    if (e >= E) return;
    const int lane = threadIdx.x & 31;
    const int s = src[e], d = dst[e];
    const float4 v = ((const float4*)(z + (size_t)s * H))[lane];
    float* o = seg + (size_t)d * H + lane * 4;
    atomicAdd(o + 0, v.x);
    atomicAdd(o + 1, v.y);
    atomicAdd(o + 2, v.z);
    atomicAdd(o + 3, v.w);
}

// seg[r][:] /= max(cnt[r], 1)
__global__ __launch_bounds__(256) void divide_kernel(
    float* __restrict__ seg, const float* __restrict__ cnt, int n) {
    const int i = blockIdx.x * blockDim.x + threadIdx.x;
    if (i < n * H) {
        const float c = cnt[i >> 7];
        seg[i] = seg[i] / fmaxf(c, 1.0f);
    }
}

// ---------------------------------------------------------------------------
// Fused hetero-SAGE update for one dst type (exactly 2 incoming relations):
// zout = 0.5*(m0@Wl0 + bl0 + m1@Wl1 + bl1 + zin@Wr0 + zin@Wr1), opt ReLU.
// Four K=128 WMMA accumulation passes into the same 16x128 panel.
// No guards in the loops: rows clamped, stores guarded.
// ---------------------------------------------------------------------------
__device__ __forceinline__ void sage_pass(
    const float* __restrict__ arow, const float* __restrict__ W,
    int half, int l16, v8f acc[8]) {
    for (int k0 = 0; k0 < H; k0 += 4) {
        const int ka = k0 + 2 * half;
        v2f a;
        a[0] = arow[ka];
        a[1] = arow[ka + 1];
        #pragma unroll
        for (int n = 0; n < 8; ++n) {
            v2f b;
            b[0] = W[ka       * H + n * 16 + l16];
            b[1] = W[(ka + 1) * H + n * 16 + l16];
            acc[n] = wmma4(a, b, acc[n]);
        }
    }
}

__global__ __launch_bounds__(256) void sage_kernel(
    const float* __restrict__ m0, const float* __restrict__ Wl0, const float* __restrict__ bl0,
    const float* __restrict__ m1, const float* __restrict__ Wl1, const float* __restrict__ bl1,
    const float* __restrict__ zin, const float* __restrict__ Wr0, const float* __restrict__ Wr1,
    float* __restrict__ zout, int M, int relu) {
    const int lane = threadIdx.x & 31;
    const int wave = threadIdx.x >> 5;
    const int half = lane >> 4;
    const int l16  = lane & 15;
    const int rowA = blockIdx.x * 128 + wave * 16 + l16;
    const size_t ro = (size_t)(rowA < M ? rowA : 0) * H;

    v8f acc[8];
    #pragma unroll
    for (int n = 0; n < 8; ++n) { v8f z = {}; acc[n] = z; }

    sage_pass(m0  + ro, Wl0, half, l16, acc);
    sage_pass(m1  + ro, Wl1, half, l16, acc);
    sage_pass(zin + ro, Wr0, half, l16, acc);
    sage_pass(zin + ro, Wr1, half, l16, acc);

    const int rbase = blockIdx.x * 128 + wave * 16 + 8 * half;
    #pragma unroll
    for (int n = 0; n < 8; ++n) {
        const int cb = n * 16 + l16;
        const float bv = 0.5f * (bl0[cb] + bl1[cb]);
        #pragma unroll
        for (int v = 0; v < 8; ++v) {
            const int r = rbase + v;
            if (r < M) {
                float o = 0.5f * acc[n][v] + bv;
                if (relu) o = fmaxf(o, 0.f);
                zout[(size_t)r * H + cb] = o;
            }
        }
    }
}

// ---------------------------------------------------------------------------
// Decoder: per pair p, h = [zcv[cvi[p]], zjob[jbi[p]]] (K=256);
// h1 = relu(h@W1 + b1); out = h1@W2 + b2.  One wave per 16 pairs; the 128-col
// h1 panel lives in 8 WMMA accumulators; K=256 splits into two clean 128-step
// halves (cv then job, no per-iteration selects); second matvec folds in-wave
// via shfl_xor reductions over the WMMA C-layout (VGPR v <-> row v / v+8).
// ---------------------------------------------------------------------------
__global__ __launch_bounds__(256) void decoder_kernel(
    const float* __restrict__ zcv, const float* __restrict__ zjob,
    const int* __restrict__ cvi, const int* __restrict__ jbi,
    const float* __restrict__ W1, const float* __restrict__ b1,
    const float* __restrict__ W2, const float* __restrict__ b2,
    float* __restrict__ out, int NP) {
    const int lane = threadIdx.x & 31;
    const int wave = threadIdx.x >> 5;
    const int half = lane >> 4;
    const int l16  = lane & 15;
    const int prow = blockIdx.x * 128 + wave * 16 + l16;
    const int pi   = prow < NP ? prow : 0;          // clamp; stores guarded below
    const float* acv = zcv  + (size_t)cvi[pi] * H;
    const float* ajb = zjob + (size_t)jbi[pi] * H;

    v8f acc[8];
    #pragma unroll
    for (int n = 0; n < 8; ++n) { v8f z = {}; acc[n] = z; }

    // first half of K: columns 0..127 of h come from z_cv (W1 rows 0..127)
    for (int k0 = 0; k0 < H; k0 += 4) {
        const int ka = k0 + 2 * half;
        v2f a;
        a[0] = acv[ka];
        a[1] = acv[ka + 1];
        #pragma unroll
        for (int n = 0; n < 8; ++n) {
            v2f b;
            b[0] = W1[(size_t)ka       * H + n * 16 + l16];
            b[1] = W1[(size_t)(ka + 1) * H + n * 16 + l16];
            acc[n] = wmma4(a, b, acc[n]);
        }
    }
    // second half of K: columns 128..255 come from z_job (W1 rows 128..255)
    const float* W1hi = W1 + (size_t)H * H;
    for (int k0 = 0; k0 < H; k0 += 4) {
        const int ka = k0 + 2 * half;
        v2f a;
        a[0] = ajb[ka];
        a[1] = ajb[ka + 1];
        #pragma unroll
        for (int n = 0; n < 8; ++n) {
            v2f b;
            b[0] = W1hi[(size_t)ka       * H + n * 16 + l16];
            b[1] = W1hi[(size_t)(ka + 1) * H + n * 16 + l16];
            acc[n] = wmma4(a, b, acc[n]);
        }
    }

    // bias + relu + weight by W2 column, sum over this lane's 8 columns
    float p[8];
    #pragma unroll
    for (int v = 0; v < 8; ++v) p[v] = 0.f;
    #pragma unroll
    for (int n = 0; n < 8; ++n) {
        const int cb = n * 16 + l16;
        const float b1v = b1[cb];
        const float w2v = W2[cb];
        #pragma unroll
        for (int v = 0; v < 8; ++v)
            p[v] += fmaxf(acc[n][v] + b1v, 0.f) * w2v;
    }
    // reduce over the 16 lanes of each half (rows v in lanes 0-15, v+8 in 16-31)
    #pragma unroll
    for (int off = 1; off < 16; off <<= 1) {
        #pragma unroll
        for (int v = 0; v < 8; ++v) p[v] += __shfl_xor(p[v], off, 32);
    }
    const float b2v = b2[0];
    if (l16 == 0) {
        const int rbase = blockIdx.x * 128 + wave * 16 + 8 * half;
        #pragma unroll
        for (int v = 0; v < 8; ++v) {
            const int r = rbase + v;
            if (r < NP) out[r] = p[v] + b2v;
        }
    }
}

// ---------------------------------------------------------------------------
// Host orchestration
// ---------------------------------------------------------------------------
extern "C" void kernel_launch(void* const* d_in, const int* in_sizes, int n_in,
                              void* d_out, int out_size, void* d_ws, size_t ws_size,
                              hipStream_t stream) {
    (void)n_in; (void)ws_size;
    // node types: 0=cv 1=job 2=skill 3=seniority
    const int NN[4] = {100000, 50000, 20000, 6};
    const int ND[4] = {386, 386, 385, 6};
    // edge types (insertion order): 0 cv->skill, 1 skill->cv, 2 job->skill,
    // 3 skill->job, 4 cv->sen, 5 sen->cv, 6 job->sen, 7 sen->job
    const int EC[8]    = {1600000, 1600000, 800000, 800000, 100000, 100000, 50000, 50000};
    const int srcOf[8] = {0, 2, 1, 2, 0, 3, 1, 3};
    const int dstOf[8] = {2, 0, 2, 1, 3, 0, 3, 1};

    const float *X[4], *PW[4], *PB[4];
    const float *Wl[2][8], *Bl[2][8], *Wr[2][8];
    const float *W1, *B1, *W2, *B2;
    const int *EDG[8], *CVI, *JBI;

    const bool sorted_order = (in_sizes[0] == 500000); // jax tree_leaves (sorted keys)
    if (!sorted_order) {
        // recursive insertion-order flattening of setup_inputs()
        for (int t = 0; t < 4; ++t) X[t] = (const float*)d_in[t];
        for (int t = 0; t < 4; ++t) { PW[t] = (const float*)d_in[4 + 2 * t];
                                      PB[t] = (const float*)d_in[5 + 2 * t]; }
        int base = 12;
        for (int l = 0; l < 2; ++l)
            for (int e = 0; e < 8; ++e) {
                Wl[l][e] = (const float*)d_in[base + 0];
                Bl[l][e] = (const float*)d_in[base + 1];
                Wr[l][e] = (const float*)d_in[base + 2];
                base += 3;
            }
        W1 = (const float*)d_in[60]; B1 = (const float*)d_in[61];
        W2 = (const float*)d_in[62]; B2 = (const float*)d_in[63];
        for (int e = 0; e < 8; ++e) EDG[e] = (const int*)d_in[64 + e];
        CVI = (const int*)d_in[72]; JBI = (const int*)d_in[73];
    } else {
        // jax pytree order: dict keys sorted at every level
        const int se[8] = {4, 0, 6, 2, 5, 7, 1, 3}; // sorted pos -> logical edge
        CVI = (const int*)d_in[0];
        for (int j = 0; j < 8; ++j) EDG[se[j]] = (const int*)d_in[1 + j];
        JBI = (const int*)d_in[9];
        W1 = (const float*)d_in[10]; W2 = (const float*)d_in[11];
        B1 = (const float*)d_in[12]; B2 = (const float*)d_in[13];
        int base = 14;                                // Wl, Wr, bl per edge
        for (int l = 0; l < 2; ++l)
            for (int j = 0; j < 8; ++j) {
                const int e = se[j];
                Wl[l][e] = (const float*)d_in[base + 0];
                Wr[l][e] = (const float*)d_in[base + 1];
                Bl[l][e] = (const float*)d_in[base + 2];
                base += 3;
            }
        const int sp[4] = {0, 1, 3, 2};               // cv, job, seniority, skill
        for (int j = 0; j < 4; ++j) { PW[sp[j]] = (const float*)d_in[62 + 2 * j];
                                      PB[sp[j]] = (const float*)d_in[63 + 2 * j]; }
        X[0] = (const float*)d_in[70]; X[1] = (const float*)d_in[71];
        X[3] = (const float*)d_in[72]; X[2] = (const float*)d_in[73];
    }

    // ---- workspace carve-out (fp32), 256B-aligned chunks ----
    float* ws = (float*)d_ws;
    size_t off = 0;
    auto take = [&](size_t nf) -> float* {
        float* p = ws + off;
        off += (nf + 63) & ~(size_t)63;
        return p;
    };
    float *zA[4], *zB[4], *SEG[8], *CNT[8];
    for (int t = 0; t < 4; ++t) zA[t] = take((size_t)NN[t] * H);
    for (int t = 0; t < 4; ++t) zB[t] = take((size_t)NN[t] * H);
    const size_t segStart = off;
    for (int e = 0; e < 8; ++e) SEG[e] = take((size_t)NN[dstOf[e]] * H);
    const size_t segFloats = off - segStart;
    float* const segBase = ws + segStart;
    const size_t cntStart = off;
    for (int e = 0; e < 8; ++e) CNT[e] = take((size_t)NN[dstOf[e]]);
    const size_t cntFloats = off - cntStart;
    float* const cntBase = ws + cntStart;

    // ---- in-degree counts (edge structure is layer-invariant) ----
    hipMemsetAsync(cntBase, 0, cntFloats * sizeof(float), stream);
    for (int e = 0; e < 8; ++e)
        count_kernel<<<dim3((EC[e] + 255) / 256), dim3(256), 0, stream>>>(
            EDG[e] + EC[e], EC[e], CNT[e]);

    // ---- input projections ----
    for (int t = 0; t < 4; ++t)
        proj_kernel<<<dim3((NN[t] + 127) / 128), dim3(256), 0, stream>>>(
            X[t], PW[t], PB[t], zA[t], NN[t], ND[t]);

    // ---- 2 hetero-SAGE layers ----
    float** zcur = zA;
    float** znxt = zB;
    const int in0[4] = {1, 3, 0, 4};  // first incoming edge type per dst
    const int in1[4] = {5, 7, 2, 6};  // second incoming edge type per dst
    for (int l = 0; l < 2; ++l) {
        hipMemsetAsync(segBase, 0, segFloats * sizeof(float), stream);
        for (int e = 0; e < 8; ++e)
            scatter_kernel<<<dim3((EC[e] + 7) / 8), dim3(256), 0, stream>>>(
                zcur[srcOf[e]], EDG[e], EDG[e] + EC[e], EC[e], SEG[e]);
        for (int e = 0; e < 8; ++e) {
            const int n = NN[dstOf[e]];
            divide_kernel<<<dim3((n * H + 255) / 256), dim3(256), 0, stream>>>(
                SEG[e], CNT[e], n);
        }
        for (int t = 0; t < 4; ++t) {
            const int e0 = in0[t], e1 = in1[t];
            sage_kernel<<<dim3((NN[t] + 127) / 128), dim3(256), 0, stream>>>(
                SEG[e0], Wl[l][e0], Bl[l][e0],
                SEG[e1], Wl[l][e1], Bl[l][e1],
                zcur[t], Wr[l][e0], Wr[l][e1],
                znxt[t], NN[t], (l == 0) ? 1 : 0);
        }
        float** tmp = zcur; zcur = znxt; znxt = tmp;
    }

    // ---- decoder over (cv, job) pairs ----
    const int NP = out_size;
    decoder_kernel<<<dim3((NP + 127) / 128), dim3(256), 0, stream>>>(
        zcur[0], zcur[1], CVI, JBI, W1, B1, W2, B2, (float*)d_out, NP);
}